// AttentionDeduplicate_26766236189058
// MI455X (gfx1250) — compile-verified
//
#include <hip/hip_runtime.h>
#include <math.h>

// ---------------------------------------------------------------------------
// MI455X (gfx1250) fused "AttentionDeduplicate" kernel.
//
// Roofline: logits GEMM = 2*B*K*Q*T*T = 17.2 GFLOP fp32 dominates; total
// memory traffic ~2MB << HBM capability at 23.3 TB/s, so compute-bound. We use
// the native fp32 matrix unit (V_WMMA_F32_16X16X4_F32, wave32) and fuse the
// swishmax + value contraction so the 268MB [B,K,Q,T] logits tensor is never
// materialized. Per-workgroup [512x128] f32 value partial lives in LDS
// (256KB of the 320KB WGP LDS -- a CDNA5-only capability). Per-key rows are
// staged LDS-side with GLOBAL_LOAD_ASYNC_TO_LDS_B32 (ASYNCcnt-tracked).
// ---------------------------------------------------------------------------

typedef float v2f __attribute__((ext_vector_type(2)));
typedef float v8f __attribute__((ext_vector_type(8)));

constexpr int B_  = 2;
constexpr int QN  = 512;
constexpr int KN  = 512;
constexpr int TD  = 128;
constexpr int NROW = B_ * KN;     // 1024 rows for [B,*,T] tensors
constexpr int KPW  = 4;           // keys per workgroup
constexpr int NWAVES = 8;         // 256 threads = 8 wave32

// ---------------- Stage 1: input projections (tiny: ~0.2 GFLOP) ------------

__global__ __launch_bounds__(256)
void proj_qk_kernel(const float* __restrict__ qtok, const float* __restrict__ ktok,
                    const float* __restrict__ Wq,   const float* __restrict__ Wk,
                    float* __restrict__ query, float* __restrict__ key,
                    float* __restrict__ vsum_zero)
{
    int idx = blockIdx.x * 256 + threadIdx.x;       // [0, NROW*TD)
    int r = idx / TD, t = idx % TD;
    const float* qrow = qtok + (size_t)r * TD;
    const float* krow = ktok + (size_t)r * TD;
    const float* wq   = Wq   + (size_t)t * TD;
    const float* wk   = Wk   + (size_t)t * TD;
    float aq = 0.f, ak = 0.f;
    #pragma unroll 4
    for (int s = 0; s < TD; ++s) {
        aq = fmaf(qrow[s], wq[s], aq);              // x @ W.T
        ak = fmaf(krow[s], wk[s], ak);
    }
    query[idx] = aq;
    key[idx]   = ak;
    vsum_zero[idx] = 0.f;                           // zero the atomic target
}

__global__ __launch_bounds__(256)
void proj_v_kernel(const float* __restrict__ key, const float* __restrict__ Wva,
                   float* __restrict__ vraw)
{
    int idx = blockIdx.x * 256 + threadIdx.x;
    int r = idx / TD, t = idx % TD;
    const float* krow = key + (size_t)r * TD;
    const float* wrow = Wva + (size_t)t * TD;
    float a = 0.f;
    #pragma unroll 4
    for (int s = 0; s < TD; ++s) a = fmaf(krow[s], wrow[s], a);
    vraw[idx] = a;
}

// ---------------- Stage 2: key norms + pairwise cosine similarity ----------

__global__ __launch_bounds__(128)
void norm_kernel(const float* __restrict__ key, float* __restrict__ inv_norm)
{
    __shared__ float red[128];
    int r = blockIdx.x, t = threadIdx.x;
    float v = key[(size_t)r * TD + t];
    red[t] = v * v;
    __syncthreads();
    for (int off = 64; off > 0; off >>= 1) {
        if (t < off) red[t] += red[t + off];
        __syncthreads();
    }
    if (t == 0) {
        float nn = fmaxf(sqrtf(red[0]), 1e-8f);     // torch cosine eps clamp
        inv_norm[r] = 1.f / nn;
    }
}

// ksim[b,k] = sum_j (kn_k . kn_j)^2   (== diag of (Kn Kn^T)^2)
__global__ __launch_bounds__(128)
void ksim_kernel(const float* __restrict__ key, const float* __restrict__ inv_norm,
                 float* __restrict__ ksim)
{
    __shared__ float kk[TD];
    __shared__ float red[128];
    int r = blockIdx.x;                 // b*KN + k
    int b = r / KN;
    int t = threadIdx.x;
    float ink = inv_norm[r];
    kk[t] = key[(size_t)r * TD + t] * ink;
    __syncthreads();
    float acc = 0.f;
    for (int j = t; j < KN; j += 128) {
        const float* krow = key + ((size_t)b * KN + j) * TD;
        float inj = inv_norm[b * KN + j];
        float dot = 0.f;
        #pragma unroll 4
        for (int s = 0; s < TD; ++s) dot = fmaf(kk[s], krow[s], dot);
        float c = dot * inj;
        acc = fmaf(c, c, acc);
    }
    red[t] = acc;
    __syncthreads();
    for (int off = 64; off > 0; off >>= 1) {
        if (t < off) red[t] += red[t + off];
        __syncthreads();
    }
    if (t == 0) ksim[r] = red[0];
}

// ---------------- Stage 3: fused logits/swishmax/value (WMMA fp32) ---------
//
// One wave computes a [16q x 128s] logits tile per q-tile:
//   acc[f] (f = s/16) accumulated over K=128 in steps of 4 via
//   V_WMMA_F32_16X16X4_F32, A[q,t] = query[q,t]*key_k[t], B[t,s] = Wal[s,t].
__device__ __forceinline__ void logits_tile(const float* __restrict__ qrow,
                                            const float* __restrict__ Wal,
                                            const float* keyrow,
                                            int hh, int n, v8f acc[8])
{
    v8f zero = {};
    #pragma unroll
    for (int f = 0; f < 8; ++f) acc[f] = zero;
    for (int t = 0; t < TD; t += 4) {
        const int ta = t + 2 * hh;                  // fp32 A frag: K = {2h, 2h+1}
        v2f a;
        a.x = qrow[ta]     * keyrow[ta];
        a.y = qrow[ta + 1] * keyrow[ta + 1];
        #pragma unroll
        for (int f = 0; f < 8; ++f) {
            const float* wr = Wal + (size_t)(f * 16 + n) * TD + ta;
            v2f bb; bb.x = wr[0]; bb.y = wr[1];     // B[t',s] = Wal[s, t+t']
            acc[f] = __builtin_amdgcn_wmma_f32_16x16x4_f32(
                false, a, false, bb, (short)0, acc[f], false, false);
        }
    }
}

__global__ __launch_bounds__(256)
void swish_attn_kernel(const float* __restrict__ query, const float* __restrict__ key,
                       const float* __restrict__ vraw,  const float* __restrict__ Wal,
                       const float* __restrict__ ksim,  float* __restrict__ vsum)
{
    extern __shared__ float smem[];
    float* vp     = smem;                   // [512][128] value partial (256 KB)
    float* st_m   = vp + QN * TD;           // [16][128] per (wave,half) run-max
    float* st_s   = st_m + 16 * TD;         // [16][128] per (wave,half) run-sum
    float* f_m    = st_s + 16 * TD;         // [128] final max per s
    float* f_d    = f_m + TD;               // [128] final denom per s
    float* keyrow = f_d + TD;               // [128]
    float* valrow = keyrow + TD;            // [128]

    const int wg   = blockIdx.x;
    const int b    = wg / (KN / KPW);
    const int k0   = (wg % (KN / KPW)) * KPW;
    const int tid  = threadIdx.x;
    const int wave = tid >> 5;              // wave32
    const int lane = tid & 31;
    const int hh   = lane >> 4;             // lane half (K / M-half select)
    const int n    = lane & 15;             // N index / A-row index

    __builtin_prefetch(Wal + (size_t)tid * 64, 0, 0);   // global_prefetch_b8

    for (int i = tid; i < QN * TD; i += 256) vp[i] = 0.f;

    const float* qb = query + (size_t)b * QN * TD;

    for (int kk = 0; kk < KPW; ++kk) {
        const int k = k0 + kk;
        __syncthreads();                    // also covers vp zeroing on iter 0

        // --- async stage per-key rows into LDS (ASYNCcnt path) ---
        // Generic pointers into LDS: addr[31:0] is the LDS byte address
        // (ISA 10.2 aperture mapping), which is what VDST must carry.
        if (tid < TD) {                     // waves 0..3: wave-uniform predicate
            const float* gk = key  + ((size_t)b * KN + k) * TD + tid;
            const float* gv = vraw + ((size_t)b * KN + k) * TD + tid;
            unsigned dk = (unsigned)(uintptr_t)(keyrow + tid);
            unsigned dv = (unsigned)(uintptr_t)(valrow + tid);
            asm volatile("global_load_async_to_lds_b32 %0, %1, off"
                         :: "v"(dk), "v"(gk) : "memory");
            asm volatile("global_load_async_to_lds_b32 %0, %1, off"
                         :: "v"(dv), "v"(gv) : "memory");
            asm volatile("s_wait_asynccnt 0x0" ::: "memory");
        }
        for (int i = tid; i < 16 * TD; i += 256) {
            st_m[i] = -INFINITY;
            st_s[i] = 0.f;
        }
        __syncthreads();

        const float kshrink = ksim[(size_t)b * KN + k];

        // ---- pass A: online column stats (max, sum |x| e^{x-max}) over q ----
        float runM[8], runS[8];
        #pragma unroll
        for (int f = 0; f < 8; ++f) { runM[f] = -INFINITY; runS[f] = 0.f; }

        for (int qt = wave; qt < QN / 16; qt += NWAVES) {
            const float* qrow = qb + (size_t)(qt * 16 + n) * TD;
            v8f acc[8];
            logits_tile(qrow, Wal, keyrow, hh, n, acc);
            #pragma unroll
            for (int f = 0; f < 8; ++f) {
                #pragma unroll
                for (int r = 0; r < 8; ++r) {
                    float x  = acc[f][r];
                    float nM = fmaxf(runM[f], x);
                    runS[f]  = runS[f] * __expf(runM[f] - nM)
                             + fabsf(x) * __expf(x - nM);
                    runM[f]  = nM;
                }
            }
        }
        {
            const int p = wave * 2 + hh;
            #pragma unroll
            for (int f = 0; f < 8; ++f) {
                st_m[p * TD + f * 16 + n] = runM[f];
                st_s[p * TD + f * 16 + n] = runS[f];
            }
        }
        __syncthreads();
        if (tid < TD) {                      // combine 16 partials per s-column
            float M = -INFINITY;
            for (int p = 0; p < 16; ++p) M = fmaxf(M, st_m[p * TD + tid]);
            float S = 0.f;
            for (int p = 0; p < 16; ++p)
                S += st_s[p * TD + tid] * __expf(st_m[p * TD + tid] - M);
            f_m[tid] = M;
            // shrink cancels: scale = x e^{x-M} / (S + ksim)
            f_d[tid] = S + kshrink;
        }
        __syncthreads();

        // ---- pass B: recompute logits, scale, accumulate value partial ----
        for (int qt = wave; qt < QN / 16; qt += NWAVES) {
            const float* qrow = qb + (size_t)(qt * 16 + n) * TD;
            v8f acc[8];
            logits_tile(qrow, Wal, keyrow, hh, n, acc);
            #pragma unroll
            for (int f = 0; f < 8; ++f) {
                const int s   = f * 16 + n;
                const float M  = f_m[s];
                const float rD = 1.f / f_d[s];
                const float vv = valrow[s];
                #pragma unroll
                for (int r = 0; r < 8; ++r) {
                    float x  = acc[f][r];
                    float sc = x * __expf(x - M) * rD;
                    vp[(size_t)(qt * 16 + r + 8 * hh) * TD + s] += vv * sc;
                }
            }
        }
    }
    __syncthreads();

    // flush private partial into global accumulator (f32 HW atomics)
    float* outp = vsum + (size_t)b * QN * TD;
    for (int i = tid; i < QN * TD; i += 256)
        unsafeAtomicAdd(&outp[i], vp[i]);
}

// ---------------- Stage 4: output projection -------------------------------

__global__ __launch_bounds__(256)
void out_proj_kernel(const float* __restrict__ vsum, const float* __restrict__ Wvo,
                     float* __restrict__ out)
{
    int idx = blockIdx.x * 256 + threadIdx.x;       // [0, B*QN*TD)
    int r = idx / TD, t = idx % TD;
    const float* vrow = vsum + (size_t)r * TD;
    const float* wrow = Wvo  + (size_t)t * TD;
    float a = 0.f;
    #pragma unroll 4
    for (int s = 0; s < TD; ++s) a = fmaf(vrow[s], wrow[s], a);
    out[idx] = a;
}

// ---------------------------------------------------------------------------

extern "C" void kernel_launch(void* const* d_in, const int* in_sizes, int n_in,
                              void* d_out, int out_size, void* d_ws, size_t ws_size,
                              hipStream_t stream)
{
    const float* qtok = (const float*)d_in[0];  // query_tokens [B,Q,T]
    const float* ktok = (const float*)d_in[1];  // key_tokens   [B,K,T]
    const float* Wk   = (const float*)d_in[2];
    const float* Wq   = (const float*)d_in[3];
    const float* Wva  = (const float*)d_in[4];
    const float* Wal  = (const float*)d_in[5];
    const float* Wvo  = (const float*)d_in[6];
    float* out = (float*)d_out;

    float* ws      = (float*)d_ws;              // ~2.1 MB used
    float* w_key   = ws;                        // [NROW*TD]
    float* w_query = w_key   + NROW * TD;
    float* w_vraw  = w_query + NROW * TD;
    float* w_vsum  = w_vraw  + NROW * TD;
    float* w_inorm = w_vsum  + NROW * TD;       // [NROW]
    float* w_ksim  = w_inorm + NROW;            // [NROW]

    dim3 blk256(256), blk128(128);
    const int nElem = NROW * TD;                // 131072

    proj_qk_kernel<<<nElem / 256, blk256, 0, stream>>>(qtok, ktok, Wq, Wk,
                                                       w_query, w_key, w_vsum);
    proj_v_kernel <<<nElem / 256, blk256, 0, stream>>>(w_key, Wva, w_vraw);
    norm_kernel   <<<NROW, blk128, 0, stream>>>(w_key, w_inorm);
    ksim_kernel   <<<NROW, blk128, 0, stream>>>(w_key, w_inorm, w_ksim);

    size_t lds_bytes = (size_t)(QN * TD + 2 * 16 * TD + 4 * TD) * sizeof(float); // 280.5 KB
    swish_attn_kernel<<<B_ * (KN / KPW), blk256, lds_bytes, stream>>>(
        w_query, w_key, w_vraw, Wal, w_ksim, w_vsum);

    out_proj_kernel<<<nElem / 256, blk256, 0, stream>>>(w_vsum, Wvo, out);
}